// AttentionIntraRNN_32787780337866
// MI455X (gfx1250) — compile-verified
//
#include <hip/hip_runtime.h>
#include <cstdint>

#define T_SEQ 512
#define BATCH 64
#define DIM   512
#define HID   512
#define G3    1536   // 3H
#define H2    1024   // 2H
#define KC    32     // K-chunk staged in LDS
#define XPAD  4      // row pad: stride 36 floats = 144B (16B-aligned, bank-spreading)

typedef __attribute__((ext_vector_type(2))) float v2f;
typedef __attribute__((ext_vector_type(8))) float v8f;

__device__ __forceinline__ v8f wmma4(v2f a, v2f b, v8f c) {
  // D = A(16x4 f32) * B(4x16 f32) + C(16x16 f32)
  return __builtin_amdgcn_wmma_f32_16x16x4_f32(false, a, false, b, (short)0, c, false, false);
}

__device__ __forceinline__ float sigf(float x) { return 1.f / (1.f + __expf(-x)); }

// CDNA5 async copy: global -> LDS, 16B per lane, tracked by ASYNCcnt.
__device__ __forceinline__ void async_ld_b128(void* lds_dst, const float* gsrc) {
  unsigned lds = (unsigned)(uintptr_t)lds_dst;            // LDS byte offset (addrspace 3)
  asm volatile("global_load_async_to_lds_b128 %0, %1, off"
               :: "v"(lds), "v"(gsrc) : "memory");
}
#define WAIT_ASYNC(n) asm volatile("s_wait_asynccnt " #n ::: "memory")

// ---------------------------------------------------------------- init
__global__ void init_kernel(const float* __restrict__ state, float* __restrict__ hbuf,
                            unsigned* __restrict__ bar) {
  int idx = blockIdx.x * blockDim.x + threadIdx.x;
  if (idx < 2 * BATCH * HID) {
    int dir = idx / (BATCH * HID);
    int rem = idx % (BATCH * HID);
    hbuf[(size_t)dir * 2 * BATCH * HID + rem] = state[idx];  // parity-0 buffer
  }
  if (idx < 2) bar[idx] = 0;
}

// ---------------------------------------------------------------- xW = gather(x) @ Wih^T + bih
// 128 threads (4 waves). 64x64 C tile, K staged in double-buffered LDS via async copies.
__global__ void xw_gemm(const int* __restrict__ embed, const float* __restrict__ lookup,
                        const float* __restrict__ WihF, const float* __restrict__ bihF,
                        const float* __restrict__ WihB, const float* __restrict__ bihB,
                        float* __restrict__ xwF, float* __restrict__ xwB) {
  const int dir = blockIdx.z;
  const float* Wih = dir ? WihB : WihF;
  const float* bih = dir ? bihB : bihF;
  float* xw = dir ? xwB : xwF;
  const int r0 = blockIdx.x * 64;   // rows of [T*B]
  const int n0 = blockIdx.y * 64;   // cols of [3H]
  __shared__ int   tokS[64];
  __shared__ float Xs[2][64][KC + XPAD];
  __shared__ float Bs[2][64][KC + XPAD];
  const int tid  = threadIdx.x;
  const int lane = tid & 31;
  const int wave = tid >> 5;        // row tile
  const int m    = lane & 15;
  const int hi   = lane >> 4;
  const int koff = hi * 2;
  if (tid < 64) tokS[tid] = embed[r0 + tid];
  __syncthreads();

  auto stage = [&](int bi, int kk) {
    // 8 async b128 per wave per chunk (4 X + 4 B)
    for (int idx = tid; idx < 64 * (KC / 4); idx += 128) {
      int row = idx >> 3, q = idx & 7;
      async_ld_b128(&Xs[bi][row][q * 4], lookup + (size_t)tokS[row] * DIM + kk + q * 4);
    }
    for (int idx = tid; idx < 64 * (KC / 4); idx += 128) {
      int row = idx >> 3, q = idx & 7;
      async_ld_b128(&Bs[bi][row][q * 4], Wih + (size_t)(n0 + row) * DIM + kk + q * 4);
    }
  };

  v8f acc[4];
#pragma unroll
  for (int ct = 0; ct < 4; ++ct) {
    float bias = bih[n0 + ct * 16 + m];
#pragma unroll
    for (int i = 0; i < 8; ++i) acc[ct][i] = bias;
  }

  stage(0, 0);
  int bi = 0;
  for (int kk = 0; kk < DIM; kk += KC, bi ^= 1) {
    if (kk + KC < DIM) { stage(bi ^ 1, kk + KC); WAIT_ASYNC(0x8); }
    else               { WAIT_ASYNC(0x0); }
    __syncthreads();
#pragma unroll
    for (int k4 = 0; k4 < KC; k4 += 4) {
      v2f a; a[0] = Xs[bi][wave*16 + m][k4 + koff]; a[1] = Xs[bi][wave*16 + m][k4 + koff + 1];
#pragma unroll
      for (int ct = 0; ct < 4; ++ct) {
        v2f b; b[0] = Bs[bi][ct*16 + m][k4 + koff]; b[1] = Bs[bi][ct*16 + m][k4 + koff + 1];
        acc[ct] = wmma4(a, b, acc[ct]);
      }
    }
    __syncthreads();
  }
#pragma unroll
  for (int ct = 0; ct < 4; ++ct) {
    int col = n0 + ct * 16 + m;
#pragma unroll
    for (int i = 0; i < 8; ++i) {
      int row = r0 + wave * 16 + i + 8 * hi;
      xw[(size_t)row * G3 + col] = acc[ct][i];
    }
  }
}

// ---------------------------------------------------------------- persistent GRU (both dirs)
// 64 blocks x 384 threads. Blocks 0..31: forward, 32..63: backward. Block owns 16 cols of H.
__global__ void gru_kernel(const float* __restrict__ WhhF, const float* __restrict__ bhhF,
                           const float* __restrict__ WhhB, const float* __restrict__ bhhB,
                           const float* __restrict__ xwF, const float* __restrict__ xwB,
                           float* __restrict__ hbuf, float* __restrict__ outbuf,
                           float* __restrict__ state_out, unsigned* __restrict__ bar) {
  const int dir = blockIdx.x >> 5;
  const int jt  = blockIdx.x & 31;
  const float* Whh = dir ? WhhB : WhhF;
  const float* bhh = dir ? bhhB : bhhF;
  const float* xw  = dir ? xwB  : xwF;
  float* hb = hbuf + (size_t)dir * 2 * BATCH * HID;
  unsigned* mybar = bar + dir;

  const int tid  = threadIdx.x;   // 384 = 12 waves
  const int lane = tid & 31;
  const int wave = tid >> 5;
  const int gate = wave % 3;      // r,z,n
  const int rt   = wave / 3;      // batch row tile 0..3
  const int m    = lane & 15;
  const int hi   = lane >> 4;
  const int koff = hi * 2;

  __shared__ float ghS[4][3][16][16];

  const int colbase = jt * 16;
  const float bias  = bhh[gate * HID + colbase + m];
  const float* wrow = Whh + (size_t)(gate * HID + colbase + m) * HID;  // B[:,n] row of Whh

  for (int t = 0; t < T_SEQ; ++t) {
    const float* hcur = hb + (size_t)(t & 1) * (BATCH * HID);
    float* hnxt       = hb + (size_t)((t + 1) & 1) * (BATCH * HID);
    const float* arow = hcur + (size_t)(rt * 16 + m) * HID;
    v8f c;
#pragma unroll
    for (int i = 0; i < 8; ++i) c[i] = bias;
#pragma unroll 4
    for (int kk = 0; kk < HID; kk += 4) {
      float2 av = *reinterpret_cast<const float2*>(arow + kk + koff);
      float2 bv = *reinterpret_cast<const float2*>(wrow + kk + koff);
      v2f a; a[0] = av.x; a[1] = av.y;
      v2f b; b[0] = bv.x; b[1] = bv.y;
      c = wmma4(a, b, c);
    }
#pragma unroll
    for (int i = 0; i < 8; ++i) ghS[rt][gate][i + 8 * hi][m] = c[i];
    __syncthreads();
    const int tx  = dir ? (T_SEQ - 1 - t) : t;
    const int txn = dir ? (tx - 1) : (tx + 1);          // next step's time index
    for (int e = tid; e < BATCH * 16; e += 384) {
      int brow = e >> 4, n = e & 15;
      int rtl = brow >> 4, mm = brow & 15;
      int col = colbase + n;
      const float* xwp = xw + ((size_t)tx * BATCH + brow) * G3;
      if (txn >= 0 && txn < T_SEQ) {                    // warm L2->WGP for step t+1
        const float* xpn = xw + ((size_t)txn * BATCH + brow) * G3;
        __builtin_prefetch(xpn + col, 0, 0);
        __builtin_prefetch(xpn + HID + col, 0, 0);
        __builtin_prefetch(xpn + 2 * HID + col, 0, 0);
      }
      float r  = sigf(xwp[col]            + ghS[rtl][0][mm][n]);
      float z  = sigf(xwp[HID + col]      + ghS[rtl][1][mm][n]);
      float nn = tanhf(xwp[2 * HID + col] + r * ghS[rtl][2][mm][n]);
      float hp = hcur[(size_t)brow * HID + col];
      float hn = (1.f - z) * nn + z * hp;
      hnxt[(size_t)brow * HID + col] = hn;
      outbuf[((size_t)tx * BATCH + brow) * H2 + dir * HID + col] = hn;
      if (t == T_SEQ - 1)
        state_out[(size_t)dir * BATCH * HID + (size_t)brow * HID + col] = hn;
    }
    __syncthreads();
    if (tid == 0) {
      __hip_atomic_fetch_add(mybar, 1u, __ATOMIC_RELEASE, __HIP_MEMORY_SCOPE_AGENT);
      unsigned target = 32u * (unsigned)(t + 1);
      while (__hip_atomic_load(mybar, __ATOMIC_ACQUIRE, __HIP_MEMORY_SCOPE_AGENT) < target)
        __builtin_amdgcn_s_sleep(2);
    }
    __syncthreads();
    __builtin_amdgcn_fence(__ATOMIC_ACQUIRE, "agent");  // invalidate WGP$: see peer h writes
  }
}

// ---------------------------------------------------------------- attn scores
// score[r] = sum_k tanh((out@W)[r,k] + b[k]) * proj[k].  64 rows/block, 8 stripes of 128 cols,
// K=1024 staged in double-buffered LDS chunks of 32 via async copies.
__global__ void attn_scores(const float* __restrict__ outbuf, const float* __restrict__ Wq,
                            const float* __restrict__ bq, const float* __restrict__ proj,
                            float* __restrict__ attnRaw) {
  const int r0   = blockIdx.x * 64;
  const int tid  = threadIdx.x;   // 256 = 8 waves
  const int lane = tid & 31;
  const int wave = tid >> 5;
  const int rt   = wave >> 1;     // row tile 0..3
  const int cp   = wave & 1;      // col half of stripe
  const int m    = lane & 15;
  const int hi   = lane >> 4;
  const int koff = hi * 2;
  __shared__ float outS[2][64][KC + XPAD];     // [buf][row][k]
  __shared__ float WS[2][KC][128 + XPAD];      // [buf][k][col]  (k-major)
  __shared__ float attnS[64];
  if (tid < 64) attnS[tid] = 0.f;
  __syncthreads();

  float part[8];
#pragma unroll
  for (int i = 0; i < 8; ++i) part[i] = 0.f;

  for (int cs = 0; cs < 8; ++cs) {             // 8 stripes x 128 cols
    auto stage = [&](int bi, int kk) {
      // 6 async b128 per wave per chunk (2 out + 4 W)
      for (int idx = tid; idx < 64 * (KC / 4); idx += 256) {
        int row = idx >> 3, q = idx & 7;
        async_ld_b128(&outS[bi][row][q * 4], outbuf + (size_t)(r0 + row) * H2 + kk + q * 4);
      }
      for (int idx = tid; idx < KC * 32; idx += 256) {
        int k = idx >> 5, q = idx & 31;
        async_ld_b128(&WS[bi][k][q * 4], Wq + (size_t)(kk + k) * H2 + cs * 128 + q * 4);
      }
    };
    v8f c[4];
#pragma unroll
    for (int j = 0; j < 4; ++j) {
      float bias = bq[cs * 128 + (cp * 4 + j) * 16 + m];
#pragma unroll
      for (int i = 0; i < 8; ++i) c[j][i] = bias;
    }
    stage(0, 0);
    int bi = 0;
    for (int kk = 0; kk < H2; kk += KC, bi ^= 1) {
      if (kk + KC < H2) { stage(bi ^ 1, kk + KC); WAIT_ASYNC(0x6); }
      else              { WAIT_ASYNC(0x0); }
      __syncthreads();
#pragma unroll
      for (int k4 = 0; k4 < KC; k4 += 4) {
        v2f a; a[0] = outS[bi][rt*16 + m][k4 + koff]; a[1] = outS[bi][rt*16 + m][k4 + koff + 1];
#pragma unroll
        for (int j = 0; j < 4; ++j) {
          int colL = (cp * 4 + j) * 16 + m;
          v2f b; b[0] = WS[bi][k4 + koff][colL]; b[1] = WS[bi][k4 + koff + 1][colL];
          c[j] = wmma4(a, b, c[j]);
        }
      }
      __syncthreads();
    }
#pragma unroll
    for (int j = 0; j < 4; ++j) {
      float pj = proj[cs * 128 + (cp * 4 + j) * 16 + m];
#pragma unroll
      for (int i = 0; i < 8; ++i) part[i] += tanhf(c[j][i]) * pj;
    }
  }
#pragma unroll
  for (int i = 0; i < 8; ++i) atomicAdd(&attnS[rt * 16 + i + 8 * hi], part[i]);
  __syncthreads();
  if (tid < 64) attnRaw[r0 + tid] = attnS[tid];
}

// ---------------------------------------------------------------- softmax over T per batch
__global__ void attn_softmax(const float* __restrict__ attnRaw, float* __restrict__ attnN,
                             float* __restrict__ d_attn_out) {
  const int b = blockIdx.x;
  const int tid = threadIdx.x;  // 128
  __shared__ float red[128];
  float v[4];
#pragma unroll
  for (int i = 0; i < 4; ++i) v[i] = attnRaw[(tid + i * 128) * BATCH + b];
  float mx = fmaxf(fmaxf(v[0], v[1]), fmaxf(v[2], v[3]));
  red[tid] = mx; __syncthreads();
  for (int s = 64; s > 0; s >>= 1) { if (tid < s) red[tid] = fmaxf(red[tid], red[tid + s]); __syncthreads(); }
  mx = red[0]; __syncthreads();
  float e[4], sum = 0.f;
#pragma unroll
  for (int i = 0; i < 4; ++i) { e[i] = __expf(v[i] - mx); sum += e[i]; }
  red[tid] = sum; __syncthreads();
  for (int s = 64; s > 0; s >>= 1) { if (tid < s) red[tid] += red[tid + s]; __syncthreads(); }
  float inv = 1.f / red[0];
#pragma unroll
  for (int i = 0; i < 4; ++i) {
    int t = tid + i * 128;
    float w = e[i] * inv;
    attnN[t * BATCH + b] = w;
    d_attn_out[b * T_SEQ + t] = w;   // attn returned as [B,T]
  }
}

// ---------------------------------------------------------------- vec[b] = sum_t attn[t,b] * out[t,b,:]
__global__ void attn_vec(const float* __restrict__ outbuf, const float* __restrict__ attnN,
                         float* __restrict__ d_vec) {
  const int b = blockIdx.x;
  const int j = blockIdx.y * 256 + threadIdx.x;
  float acc = 0.f;
  for (int t = 0; t < T_SEQ; ++t)
    acc += attnN[t * BATCH + b] * outbuf[((size_t)t * BATCH + b) * H2 + j];
  d_vec[b * H2 + j] = acc;
}

// ---------------------------------------------------------------- launch
extern "C" void kernel_launch(void* const* d_in, const int* in_sizes, int n_in,
                              void* d_out, int out_size, void* d_ws, size_t ws_size,
                              hipStream_t stream) {
  const int*   embed  = (const int*)d_in[0];
  const float* state  = (const float*)d_in[1];
  const float* lookup = (const float*)d_in[2];
  const float* WihF   = (const float*)d_in[3];
  const float* WhhF   = (const float*)d_in[4];
  const float* bihF   = (const float*)d_in[5];
  const float* bhhF   = (const float*)d_in[6];
  const float* WihB   = (const float*)d_in[7];
  const float* WhhB   = (const float*)d_in[8];
  const float* bihB   = (const float*)d_in[9];
  const float* bhhB   = (const float*)d_in[10];
  const float* Wq     = (const float*)d_in[11];
  const float* bq     = (const float*)d_in[12];
  const float* proj   = (const float*)d_in[13];

  float* out       = (float*)d_out;
  float* vec_out   = out;                 // [64][1024]
  float* state_out = out + 65536;         // [2][64][512]
  float* attn_out  = out + 131072;        // [64][512]

  float* ws = (float*)d_ws;
  const size_t XW = (size_t)T_SEQ * BATCH * G3;          // 50,331,648 f32
  float* xwF     = ws;
  float* xwB     = ws + XW;
  float* outbuf  = ws + 2 * XW;                          // [T][B][2H]
  float* hbuf    = outbuf + (size_t)T_SEQ * BATCH * H2;  // [2 dir][2 parity][B][H]
  float* attnRaw = hbuf + 2 * 2 * BATCH * HID;           // [T][B]
  float* attnN   = attnRaw + T_SEQ * BATCH;              // [T][B]
  unsigned* bar  = (unsigned*)(attnN + T_SEQ * BATCH);   // [2]

  init_kernel<<<dim3(256), dim3(256), 0, stream>>>(state, hbuf, bar);
  xw_gemm<<<dim3(512, 24, 2), dim3(128), 0, stream>>>(embed, lookup, WihF, bihF, WihB, bihB, xwF, xwB);
  gru_kernel<<<dim3(64), dim3(384), 0, stream>>>(WhhF, bhhF, WhhB, bhhB, xwF, xwB,
                                                 hbuf, outbuf, state_out, bar);
  attn_scores<<<dim3(512), dim3(256), 0, stream>>>(outbuf, Wq, bq, proj, attnRaw);
  attn_softmax<<<dim3(64), dim3(128), 0, stream>>>(attnRaw, attnN, attn_out);
  attn_vec<<<dim3(64, 4), dim3(256), 0, stream>>>(outbuf, attnN, vec_out);
}